// FilteredNoiseGenerator_609885356329
// MI455X (gfx1250) — compile-verified
//
#include <hip/hip_runtime.h>
#include <math.h>

typedef __attribute__((ext_vector_type(2))) float v2f;
typedef __attribute__((ext_vector_type(8))) float v8f;

#define NBANDS 65
#define LFIR   129
#define NPAD   144         // 129 padded to 9 tiles of 16
#define FRAME  80
#define TFRAMES 4000
#define BATCH  32
#define LCONV  208         // FRAME + LFIR - 1
#define ROWS   (BATCH*TFRAMES)   // 128000
#define NTILES 9
#define RTILES (ROWS/16)         // 8000
#define KSTEPS 17                // ceil(65/4)
#define PPAIRS (KSTEPS*2)        // 34 (K-pairs: k0 = 2p)

// ---------------------------------------------------------------------------
// Kernel 1: build the fused (irfft + roll + hann) coefficient table, packed
// in exactly the B-fragment order the WMMA loop consumes:
//   Bp[(p*NPAD + n)*2 + r] = Mcoef(k = 2p + r, n)
//   Mcoef(k,n) = hann(n) * w_k * cos(2*pi*k*(n-64)/129) / 129
// zero for k >= 65 or n >= 129.  Pair (2p, 2p+1) is contiguous -> b64 loads.
// ---------------------------------------------------------------------------
__global__ void build_B(float* __restrict__ Bp) {
  int i = blockIdx.x * blockDim.x + threadIdx.x;
  if (i >= PPAIRS * NPAD * 2) return;
  int p   = i / (NPAD * 2);
  int rem = i - p * (NPAD * 2);
  int n   = rem >> 1;
  int r   = rem & 1;
  int k   = 2 * p + r;
  float v = 0.0f;
  if (k < NBANDS && n < LFIR) {
    const float PI = 3.14159265358979323846f;
    float hann = 0.5f * (1.0f - cosf(2.0f * PI * (float)n / (float)LFIR));
    float w = (k == 0) ? 1.0f : 2.0f;
    float c = cosf(2.0f * PI * (float)k * (float)(n - (NBANDS - 1)) / (float)LFIR);
    v = hann * w * c / (float)LFIR;
  }
  Bp[i] = v;
}

// ---------------------------------------------------------------------------
// Kernel 2: FIR[row, n] = sum_k H[row, k] * Mcoef(k, n)  via
// V_WMMA_F32_16X16X4_F32.  One wave computes a full 16x144 strip: the A
// fragment (16x4 of H) is loaded once per k-step and reused by 9 WMMAs into
// 9 independent accumulators (no D->A/B hazard between them).
// Main loop kk=0..15 is branch-free (k <= 63 always valid); the kk=16 tail
// only needs k=64 (B rows 65..67 are zero), handled with one v_cndmask.
// ---------------------------------------------------------------------------
__global__ void fir_gemm(const float* __restrict__ H,
                         const float* __restrict__ Bp,
                         float* __restrict__ FIR) {
  int wave = (int)((blockIdx.x * blockDim.x + threadIdx.x) >> 5);
  int lane = (int)(threadIdx.x & 31u);
  if (wave >= RTILES) return;                // wave-uniform guard

  int half = lane >> 4;                      // K-half select within fragment
  int lm   = lane & 15;                      // A: M row; B: N column

  int row = wave * 16 + lm;                  // < 128000 always
  const float* hrow = H + (size_t)row * NBANDS;

  v8f c0 = {}, c1 = {}, c2 = {}, c3 = {}, c4 = {}, c5 = {}, c6 = {}, c7 = {}, c8 = {};

#pragma unroll 1
  for (int kk = 0; kk < KSTEPS - 1; ++kk) {
    int kA = kk * 4 + 2 * half;              // <= 62, kA+1 <= 63 < 65: no guards
    v2f a;
    a.x = hrow[kA];
    a.y = hrow[kA + 1];
    const v2f* brow = (const v2f*)(Bp + ((size_t)(kk * 2 + half) * NPAD + lm) * 2);
    v2f b0 = brow[0],  b1 = brow[16],  b2 = brow[32],  b3 = brow[48],
        b4 = brow[64], b5 = brow[80],  b6 = brow[96],  b7 = brow[112],
        b8 = brow[128];
    c0 = __builtin_amdgcn_wmma_f32_16x16x4_f32(false, a, false, b0, (short)0, c0, false, false);
    c1 = __builtin_amdgcn_wmma_f32_16x16x4_f32(false, a, false, b1, (short)0, c1, false, false);
    c2 = __builtin_amdgcn_wmma_f32_16x16x4_f32(false, a, false, b2, (short)0, c2, false, false);
    c3 = __builtin_amdgcn_wmma_f32_16x16x4_f32(false, a, false, b3, (short)0, c3, false, false);
    c4 = __builtin_amdgcn_wmma_f32_16x16x4_f32(false, a, false, b4, (short)0, c4, false, false);
    c5 = __builtin_amdgcn_wmma_f32_16x16x4_f32(false, a, false, b5, (short)0, c5, false, false);
    c6 = __builtin_amdgcn_wmma_f32_16x16x4_f32(false, a, false, b6, (short)0, c6, false, false);
    c7 = __builtin_amdgcn_wmma_f32_16x16x4_f32(false, a, false, b7, (short)0, c7, false, false);
    c8 = __builtin_amdgcn_wmma_f32_16x16x4_f32(false, a, false, b8, (short)0, c8, false, false);
  }

  // Tail k-step (kk = 16): k = {64,65} / {66,67}. Only k=64 contributes
  // (Bp rows for k>=65 are zero-filled); select is branch-free.
  {
    float h64 = hrow[NBANDS - 1];
    v2f a;
    a.x = half ? 0.0f : h64;
    a.y = 0.0f;
    const int kk = KSTEPS - 1;
    const v2f* brow = (const v2f*)(Bp + ((size_t)(kk * 2 + half) * NPAD + lm) * 2);
    v2f b0 = brow[0],  b1 = brow[16],  b2 = brow[32],  b3 = brow[48],
        b4 = brow[64], b5 = brow[80],  b6 = brow[96],  b7 = brow[112],
        b8 = brow[128];
    c0 = __builtin_amdgcn_wmma_f32_16x16x4_f32(false, a, false, b0, (short)0, c0, false, false);
    c1 = __builtin_amdgcn_wmma_f32_16x16x4_f32(false, a, false, b1, (short)0, c1, false, false);
    c2 = __builtin_amdgcn_wmma_f32_16x16x4_f32(false, a, false, b2, (short)0, c2, false, false);
    c3 = __builtin_amdgcn_wmma_f32_16x16x4_f32(false, a, false, b3, (short)0, c3, false, false);
    c4 = __builtin_amdgcn_wmma_f32_16x16x4_f32(false, a, false, b4, (short)0, c4, false, false);
    c5 = __builtin_amdgcn_wmma_f32_16x16x4_f32(false, a, false, b5, (short)0, c5, false, false);
    c6 = __builtin_amdgcn_wmma_f32_16x16x4_f32(false, a, false, b6, (short)0, c6, false, false);
    c7 = __builtin_amdgcn_wmma_f32_16x16x4_f32(false, a, false, b7, (short)0, c7, false, false);
    c8 = __builtin_amdgcn_wmma_f32_16x16x4_f32(false, a, false, b8, (short)0, c8, false, false);
  }

  // C/D layout: vgpr v, lane-half h -> output row (v + 8h), col = lm
  size_t rbase = (size_t)(wave * 16 + 8 * half) * NPAD;
  v8f acc[9] = {c0, c1, c2, c3, c4, c5, c6, c7, c8};
#pragma unroll
  for (int j = 0; j < NTILES; ++j) {
    size_t outBase = rbase + j * 16 + lm;
#pragma unroll
    for (int v = 0; v < 8; ++v)
      FIR[outBase + (size_t)v * NPAD] = acc[j][v];
  }
}

// ---------------------------------------------------------------------------
// Kernel 3: time-domain convolution + overlap-add + group-delay trim.
// One thread per trimmed output sample:
//   y[b,s] = sum over frames t (<=3) and taps k of (2*noise-1)*fir
// Loop over tap k so fir reads are frame-uniform and noise reads coalesce.
// ---------------------------------------------------------------------------
__global__ void conv_ola(const float* __restrict__ noise,
                         const float* __restrict__ FIR,
                         float* __restrict__ out) {
  long long idx = (long long)blockIdx.x * blockDim.x + threadIdx.x;
  if (idx >= (long long)BATCH * TFRAMES * FRAME) return;
  int b = (int)(idx / (TFRAMES * FRAME));
  int s = (int)(idx - (long long)b * (TFRAMES * FRAME));

  int u = s + (NBANDS - 1);                  // position in untrimmed OLA buffer
  int t_hi = u / FRAME; if (t_hi > TFRAMES - 1) t_hi = TFRAMES - 1;
  int lo = u - (LCONV - 1);
  int t_lo = (lo <= 0) ? 0 : (lo + FRAME - 1) / FRAME;

  float acc = 0.0f;
  for (int t = t_lo; t <= t_hi; ++t) {
    int n = u - t * FRAME;                   // 0..207, valid by construction
    int kmin = n - (FRAME - 1); if (kmin < 0) kmin = 0;
    int kmax = n;               if (kmax > LFIR - 1) kmax = LFIR - 1;
    const float* xr = noise + (size_t)(b * TFRAMES + t) * FRAME;
    const float* fr = FIR   + (size_t)(b * TFRAMES + t) * NPAD;
    for (int k = kmin; k <= kmax; ++k) {
      float x = 2.0f * xr[n - k] - 1.0f;
      acc = fmaf(x, fr[k], acc);
    }
  }
  out[idx] = acc;
}

// ---------------------------------------------------------------------------
extern "C" void kernel_launch(void* const* d_in, const int* in_sizes, int n_in,
                              void* d_out, int out_size, void* d_ws, size_t ws_size,
                              hipStream_t stream) {
  const float* H     = (const float*)d_in[0];   // [32, 4000, 65]
  const float* noise = (const float*)d_in[1];   // [32, 4000, 80]
  // d_in[2] = framesize scalar (80), compile-time constant here.
  float* out = (float*)d_out;                   // [32, 320000]

  float* Bp  = (float*)d_ws;                              // 34*144*2 floats
  float* FIR = (float*)((char*)d_ws + 64 * 1024);         // 128000*144 floats

  {
    int total = PPAIRS * NPAD * 2;                        // 9792
    build_B<<<(total + 255) / 256, 256, 0, stream>>>(Bp);
  }
  {
    long long threads = (long long)RTILES * 32;           // 8000 waves
    fir_gemm<<<(unsigned)((threads + 255) / 256), 256, 0, stream>>>(H, Bp, FIR);
  }
  {
    long long total = (long long)BATCH * TFRAMES * FRAME; // 10,240,000
    conv_ola<<<(unsigned)((total + 255) / 256), 256, 0, stream>>>(noise, FIR, out);
  }
}